// GraphLSTMEncoder_43258910605945
// MI455X (gfx1250) — compile-verified
//
#include <hip/hip_runtime.h>
#include <math.h>

// Problem constants (from reference setup_inputs)
#define B_    16
#define S_    24
#define N_    1024
#define D_    8
#define H_    64
#define L_    2
#define TWOH  128
#define FOURH 256

typedef __attribute__((ext_vector_type(2))) float v2f;
typedef __attribute__((ext_vector_type(8))) float v8f;

// ---- LDS tiling for the fused graph-conv + gates kernel -------------------
// adj tile   : 64 rows x 64 k      row-major, pitch 66 (pad, keeps b64 align)
// featT tile : 128 d    x 64 n     (transposed feat), pitch 66
// sup tile   : 64 rows x 128 cols  row-major, pitch 130
// WgT        : 256 cols x 128 k    (transposed W_g), pitch 130
#define KB          64
#define ADJ_PITCH   66
#define FEATT_PITCH 66
#define SUP_PITCH   130
#define WG_PITCH    130
#define LDS_ADJ_OFF   0
#define LDS_FEATT_OFF (64 * ADJ_PITCH)                     // 4224
#define LDS_SUP_OFF   (LDS_FEATT_OFF + TWOH * FEATT_PITCH) // 12672
#define LDS_WG_OFF    (LDS_SUP_OFF + 64 * SUP_PITCH)       // 20992
#define LDS_TOTAL_F   (LDS_WG_OFF + FOURH * WG_PITCH)      // 54272 floats
#define LDS_BYTES     (LDS_TOTAL_F * 4)                    // 217088 B < 320KB WGP LDS

__device__ __forceinline__ float sigm(float v) {
    return 1.0f / (1.0f + __expf(-v));
}

__device__ __forceinline__ v8f wmma_f32(v2f a, v2f b, v8f c) {
    // D = A(16x4 f32) * B(4x16 f32) + C(16x16 f32), full fp32 matrix pipe
    return __builtin_amdgcn_wmma_f32_16x16x4_f32(
        /*neg_a=*/false, a, /*neg_b=*/false, b,
        /*c_mod=*/(short)0, c, /*reuse_a=*/false, /*reuse_b=*/false);
}

// ---------------------------------------------------------------------------
// Workspace zero-init (h and c state must start at 0; harness poisons ws)
__global__ void zero_f32(float* __restrict__ p, size_t n) {
    size_t i = (size_t)blockIdx.x * blockDim.x + threadIdx.x;
    size_t stride = (size_t)gridDim.x * blockDim.x;
    for (; i < n; i += stride) p[i] = 0.0f;
}

// ---------------------------------------------------------------------------
// Row-normalized adjacency: adj_n[m][n] = adj[m][n] / (sum_n adj[m][n] + 1e-8)
__global__ __launch_bounds__(256) void adj_norm(const float* __restrict__ adj,
                                                float* __restrict__ adjn) {
    __shared__ float red[256];
    const int row = blockIdx.x;
    const int tid = threadIdx.x;
    float s = 0.0f;
    for (int c = tid; c < N_; c += 256) s += adj[(size_t)row * N_ + c];
    red[tid] = s;
    __syncthreads();
    for (int off = 128; off > 0; off >>= 1) {
        if (tid < off) red[tid] += red[tid + off];
        __syncthreads();
    }
    const float inv = 1.0f / (red[0] + 1e-8f);
    for (int c = tid; c < N_; c += 256)
        adjn[(size_t)row * N_ + c] = adj[(size_t)row * N_ + c] * inv;
}

// ---------------------------------------------------------------------------
// Per layer-step prep: builds feat[b][n][0:64]   = LN(layer_in)+node_embed
//                             feat[b][n][64:128] = h_old[layer]
// layer 0: layer_in = x[:,t] @ W_in + b_in + PE(t)   (fused tiny GEMM)
// layer 1: layer_in = h[0] (freshly updated by the previous step kernel)
// One wave32 per (b, n); LN reduction via lane shuffles.
__global__ __launch_bounds__(256) void feat_prep(
    int layer, int t,
    const float* __restrict__ x, const float* __restrict__ W_in,
    const float* __restrict__ b_in, const float* __restrict__ node_embed,
    const float* __restrict__ ln_g, const float* __restrict__ ln_b,
    const float* __restrict__ h_all, float* __restrict__ feat) {
    const int gw = (blockIdx.x * blockDim.x + threadIdx.x) >> 5; // wave id = (b,n)
    const int l = threadIdx.x & 31;
    const int b = gw >> 10;
    const int n = gw & (N_ - 1);

    float v0, v1; // this lane's two H-slots: j = l and j = l+32
    if (layer == 0) {
        float xv[D_];
        const float* xp = x + (((size_t)b * S_ + t) * N_ + n) * D_;
#pragma unroll
        for (int d = 0; d < D_; ++d) xv[d] = xp[d];
#pragma unroll
        for (int e = 0; e < 2; ++e) {
            const int j = l + 32 * e;
            float a = b_in[j];
#pragma unroll
            for (int d = 0; d < D_; ++d) a += xv[d] * W_in[d * H_ + j];
            // sinusoidal PE: freq = 10000^{-(j&~1)/H}
            const float freq = __expf((float)(j & ~1) * (-0.1439115683f)); // ln(1e4)/64
            const float arg = (float)t * freq;
            a += (j & 1) ? __cosf(arg) : __sinf(arg);
            if (e == 0) v0 = a; else v1 = a;
        }
    } else {
        const size_t base = ((size_t)(b * N_ + n)) * H_;
        v0 = h_all[base + l];        // layer 0 h == layer_in
        v1 = h_all[base + l + 32];
    }

    // LayerNorm over H=64 (population variance, eps=1e-5)
    float s = v0 + v1, ss = v0 * v0 + v1 * v1;
#pragma unroll
    for (int m = 16; m >= 1; m >>= 1) {
        s += __shfl_xor(s, m, 32);
        ss += __shfl_xor(ss, m, 32);
    }
    const float mean = s * (1.0f / H_);
    const float var = ss * (1.0f / H_) - mean * mean;
    const float rstd = rsqrtf(var + 1e-5f);

    const float* hold = h_all + (size_t)layer * B_ * N_ * H_;
    const size_t base = (size_t)(b * N_ + n);
#pragma unroll
    for (int e = 0; e < 2; ++e) {
        const int j = l + 32 * e;
        const float vv = (e == 0) ? v0 : v1;
        const float z = (vv - mean) * rstd * ln_g[layer * H_ + j] +
                        ln_b[layer * H_ + j] + node_embed[n * H_ + j];
        feat[base * TWOH + j] = z;
        feat[base * TWOH + H_ + j] = hold[base * H_ + j];
    }
}

// ---------------------------------------------------------------------------
// Fused: sup = adj_n @ feat  ->  gates = sup @ W_g + b_g  ->  LSTM pointwise.
// Grid (N/64, B); 128 threads = 4 waves; wave wr owns sup rows [wr*16, wr*16+16).
// All intermediates (sup tile, W_gT) live in 212KB of LDS (CDNA5 WGP has 320KB).
__global__ __launch_bounds__(128) void graph_lstm_step(
    const float* __restrict__ adjn, const float* __restrict__ feat,
    const float* __restrict__ Wg, const float* __restrict__ bg,
    float* __restrict__ cbuf, float* __restrict__ hbuf,
    const float* __restrict__ resid, float* __restrict__ outp, int t) {
    extern __shared__ float smem[];
    float* lds_adj = smem + LDS_ADJ_OFF;
    float* lds_featT = smem + LDS_FEATT_OFF;
    float* lds_sup = smem + LDS_SUP_OFF;
    float* lds_wgT = smem + LDS_WG_OFF;

    const int tid = threadIdx.x;
    const int wr = tid >> 5;       // wave in WG (0..3)
    const int l = tid & 31;        // lane
    const int half = l >> 4;       // lane half (K-pair select in WMMA frags)
    const int lan = l & 15;        // lane within half
    const int rowBase = blockIdx.x * 64;
    const int b = blockIdx.y;

    // Stage W_g transposed into LDS once: lds_wgT[col][k] = Wg[k][col]
    for (int idx = tid; idx < TWOH * FOURH; idx += 128) {
        const int k = idx >> 8;      // / 256
        const int col = idx & 255;
        lds_wgT[col * WG_PITCH + k] = Wg[idx];
    }

    // ---- Stage 1: sup[rowBase..+64][0..128) = adj_n[rows] @ feat[b] --------
    v8f acc[8] = {}; // 8 x (16x16) col tiles covering cols 0..127
    const float* adjFrag = lds_adj + (wr * 16 + lan) * ADJ_PITCH + 2 * half;

    for (int kb = 0; kb < N_; kb += KB) {
        __syncthreads();
        // adj tile 64x64 (row-major), float4 global loads
#pragma unroll
        for (int it = 0; it < 8; ++it) {
            const int q = it * 128 + tid;      // float4 index 0..1023
            const int r = q >> 4;
            const int c4 = (q & 15) << 2;
            const float4 vv =
                *reinterpret_cast<const float4*>(adjn + (size_t)(rowBase + r) * N_ + kb + c4);
            float* sp = lds_adj + r * ADJ_PITCH + c4;
            sp[0] = vv.x; sp[1] = vv.y; sp[2] = vv.z; sp[3] = vv.w;
        }
        // feat tile transposed: lds_featT[d][n] = feat[b][kb+n][d] (coalesced reads)
        if (kb + KB < N_)
            __builtin_prefetch(feat + ((size_t)(b * N_ + kb + KB) + (tid >> 1)) * TWOH, 0, 0);
#pragma unroll 4
        for (int n = 0; n < KB; ++n) {
            lds_featT[tid * FEATT_PITCH + n] = feat[((size_t)(b * N_ + kb + n)) * TWOH + tid];
        }
        __syncthreads();

#pragma unroll 4
        for (int kk = 0; kk < KB; kk += 4) {
            const v2f a = *reinterpret_cast<const v2f*>(adjFrag + kk); // A frag (16x4)
#pragma unroll
            for (int ct = 0; ct < 8; ++ct) {
                const v2f bf = *reinterpret_cast<const v2f*>(
                    lds_featT + (ct * 16 + lan) * FEATT_PITCH + kk + 2 * half); // B frag (4x16)
                acc[ct] = wmma_f32(a, bf, acc[ct]);
            }
        }
    }

    // Park sup tile in LDS (row-major) so stage 2 can re-read it K-major.
    __syncthreads();
#pragma unroll
    for (int ct = 0; ct < 8; ++ct) {
#pragma unroll
        for (int r = 0; r < 8; ++r) {
            lds_sup[(wr * 16 + r + 8 * half) * SUP_PITCH + ct * 16 + lan] = acc[ct][r];
        }
    }
    __syncthreads();

    // ---- Stage 2: gates = sup @ W_g  (K=128, 16 col tiles over 256 cols) ---
    v8f g[16] = {};
    const float* supFrag = lds_sup + (wr * 16 + lan) * SUP_PITCH + 2 * half;
#pragma unroll 4
    for (int kk = 0; kk < TWOH; kk += 4) {
        const v2f a = *reinterpret_cast<const v2f*>(supFrag + kk);
#pragma unroll
        for (int ct = 0; ct < 16; ++ct) {
            const v2f bf = *reinterpret_cast<const v2f*>(
                lds_wgT + (ct * 16 + lan) * WG_PITCH + kk + 2 * half);
            g[ct] = wmma_f32(a, bf, g[ct]);
        }
    }

    // ---- LSTM pointwise on register-resident gate tiles --------------------
    // gates cols: [0,64)=ig  [64,128)=fg  [128,192)=gg  [192,256)=og
#pragma unroll
    for (int jt = 0; jt < 4; ++jt) {
        const float big = bg[jt * 16 + lan];
        const float bfg = bg[H_ + jt * 16 + lan];
        const float bgg = bg[2 * H_ + jt * 16 + lan];
        const float bog = bg[3 * H_ + jt * 16 + lan];
#pragma unroll
        for (int r = 0; r < 8; ++r) {
            const int row = wr * 16 + r + 8 * half;
            const int n = rowBase + row;
            const int j = jt * 16 + lan;
            const size_t idx = ((size_t)(b * N_ + n)) * H_ + j;
            const float ig = g[jt][r] + big;
            const float fg = g[jt + 4][r] + bfg;
            const float gv = g[jt + 8][r] + bgg;
            const float og = g[jt + 12][r] + bog;
            const float cold = cbuf[idx];
            const float cnew = sigm(fg) * cold + sigm(ig) * tanhf(gv);
            float hnew = sigm(og) * tanhf(cnew);
            if (resid) hnew += resid[idx];     // residual: layer_in (h of layer 0)
            cbuf[idx] = cnew;
            hbuf[idx] = hnew;
            if (outp) outp[(((size_t)b * S_ + t) * N_ + n) * H_ + j] = hnew;
        }
    }
}

// ---------------------------------------------------------------------------
extern "C" void kernel_launch(void* const* d_in, const int* in_sizes, int n_in,
                              void* d_out, int out_size, void* d_ws, size_t ws_size,
                              hipStream_t stream) {
    (void)in_sizes; (void)n_in; (void)out_size; (void)ws_size;
    const float* x = (const float*)d_in[0];
    const float* adj = (const float*)d_in[1];
    const float* W_in = (const float*)d_in[2];
    const float* b_in = (const float*)d_in[3];
    const float* node_embed = (const float*)d_in[4];
    const float* ln_g = (const float*)d_in[5];
    const float* ln_b = (const float*)d_in[6];
    const float* W_g = (const float*)d_in[7];
    const float* b_g = (const float*)d_in[8];
    float* out = (float*)d_out;

    // Workspace layout (floats):
    float* ws = (float*)d_ws;
    float* adjn = ws;                                   // 1024*1024      = 1048576
    float* hbuf = adjn + (size_t)N_ * N_;               // L*B*N*H        = 2097152
    float* cbuf = hbuf + (size_t)L_ * B_ * N_ * H_;     // L*B*N*H        = 2097152
    float* feat = cbuf + (size_t)L_ * B_ * N_ * H_;     // B*N*2H         = 2097152

    // Allow >64KB dynamic LDS for the fused kernel (CDNA5 WGP: 320KB).
    hipFuncSetAttribute(reinterpret_cast<const void*>(graph_lstm_step),
                        hipFuncAttributeMaxDynamicSharedMemorySize, LDS_BYTES);

    // h0 = c0 = 0 (contiguous h then c regions)
    zero_f32<<<2048, 256, 0, stream>>>(hbuf, (size_t)2 * L_ * B_ * N_ * H_);
    adj_norm<<<N_, 256, 0, stream>>>(adj, adjn);

    for (int t = 0; t < S_; ++t) {
        for (int layer = 0; layer < L_; ++layer) {
            feat_prep<<<(B_ * N_) / 8, 256, 0, stream>>>(
                layer, t, x, W_in, b_in, node_embed, ln_g, ln_b, hbuf, feat);
            graph_lstm_step<<<dim3(N_ / 64, B_), 128, LDS_BYTES, stream>>>(
                adjn, feat,
                W_g + (size_t)layer * TWOH * FOURH, b_g + (size_t)layer * FOURH,
                cbuf + (size_t)layer * B_ * N_ * H_, hbuf + (size_t)layer * B_ * N_ * H_,
                (layer > 0) ? hbuf : nullptr,
                (layer == L_ - 1) ? out : nullptr, t);
        }
    }
}